// MultiSineGaussian_16741782519952
// MI455X (gfx1250) — compile-verified
//
#include <hip/hip_runtime.h>
#include <math.h>

typedef __attribute__((ext_vector_type(2))) float v2f;
typedef __attribute__((ext_vector_type(8))) float v8f;

#define T_LEN 16384
#define G_CNT 64
#define N_CNT 32
#define T_PER_BLOCK 128   // 8 waves * 16 t each

__global__ __launch_bounds__(256) void msg_wmma_kernel(
    const float* __restrict__ quality,
    const float* __restrict__ frequency,
    const float* __restrict__ hrss,
    const float* __restrict__ phase,
    const float* __restrict__ ecc,
    const float* __restrict__ shifts,
    float* __restrict__ outCross,
    float* __restrict__ outPlus)
{
    __shared__ float sTwoPiF[N_CNT];
    __shared__ float sP[N_CNT];
    __shared__ float sH0c[N_CNT];
    __shared__ float sH0p[N_CNT];
    __shared__ float sNegInv2q2[N_CNT];
    __shared__ int   sShift[N_CNT];
    __shared__ float stage[8][2][16];

    const int g   = blockIdx.y;
    const int tid = threadIdx.x;

    // ---- per-component derived constants (once per block) ----
    if (tid < N_CNT) {
        const int n = tid;
        const int i = g * N_CNT + n;
        const float q = quality[i];
        const float f = frequency[i];
        const float h = hrss[i];
        const float p = phase[i];
        const float e = ecc[i];

        const float a    = 1.0f / sqrtf(2.0f - e * e);
        const float bmin = a * sqrtf(1.0f - e * e);

        const float pre  = q / (4.0f * f * 1.7724538509055160273f); // sqrt(pi)
        const float expq = expf(-q * q);
        const float cn   = pre * (1.0f + expq);
        const float sn   = pre * (1.0f - expq);

        float sp, cp;
        sincosf(p, &sp, &cp);

        sH0p[n]       = h * a    / sqrtf(cn * cp * cp + sn * sp * sp);
        sH0c[n]       = h * bmin / sqrtf(cn * sp * sp + sn * cp * cp);
        sTwoPiF[n]    = 6.28318530717958647692f * f;
        sP[n]         = p;
        sNegInv2q2[n] = -1.0f / (2.0f * q * q);
        sShift[n]     = (int)(shifts[i] * 4096.0f);  // trunc toward zero, matches astype(int32)
    }
    __syncthreads();

    const int lane = tid & 31;
    const int wave = tid >> 5;
    const int m    = lane & 15;          // row M of A, column N of D
    const int kOff = (lane >> 4) << 1;   // K offset: lanes 0-15 -> {0,1}, 16-31 -> {2,3}
    const int tBase = blockIdx.x * T_PER_BLOCK + wave * 16;
    const int t     = tBase + m;

    v8f dC = {};   // cross accumulator (16x16 f32 D, columns all equal by construction)
    v8f dP = {};   // plus accumulator
    v2f ones; ones[0] = 1.0f; ones[1] = 1.0f;  // B = all-ones 4x16 (layout-invariant)

    // Rolled loop: keeps the hot transcendental body (2 sincosf + 2 v_exp)
    // small enough to stay resident in the WGP I$; WMMAs accumulate in-loop.
    #pragma unroll 1
    for (int c = 0; c < 8; ++c) {        // sweep N = 32 in K=4 chunks
        v2f aC, aP;
        #pragma unroll
        for (int k = 0; k < 2; ++k) {
            const int n   = (c << 2) + kOff + k;
            const int idx = t - sShift[n];
            const float vm = (idx >= 0 && idx < T_LEN) ? 1.0f : 0.0f;

            const float time = (float)idx * (1.0f / 4096.0f) - 2.0f;
            const float phi  = sTwoPiF[n] * time;
            const float env  = __expf(phi * phi * sNegInv2q2[n]);  // v_exp_f32 path
            float s, cc;
            sincosf(phi - sP[n], &s, &cc);  // accurate reduction; |arg| < 2^15 keeps fast path

            aC[k] = vm * env * s  * sH0c[n];
            aP[k] = vm * env * cc * sH0p[n];
        }
        // D += A(16x4, f32) x B(4x16, ones) : full-precision reduction over 4 components
        dC = __builtin_amdgcn_wmma_f32_16x16x4_f32(false, aC, false, ones,
                                                   (short)0, dC, false, false);
        dP = __builtin_amdgcn_wmma_f32_16x16x4_f32(false, aP, false, ones,
                                                   (short)0, dP, false, false);
    }

    // ---- extract column 0 of each D via LDS (lane 0 holds rows 0-7, lane 16 rows 8-15) ----
    if (lane == 0 || lane == 16) {
        const int rbase = (lane == 0) ? 0 : 8;
        #pragma unroll
        for (int r = 0; r < 8; ++r) {
            stage[wave][0][rbase + r] = dC[r] * (1.0f / 32.0f);  // mean over N
            stage[wave][1][rbase + r] = dP[r] * (1.0f / 32.0f);
        }
    }
    __syncthreads();

    if (lane < 16) {
        outCross[g * T_LEN + tBase + lane] = stage[wave][0][lane];
    } else {
        outPlus[g * T_LEN + tBase + (lane - 16)] = stage[wave][1][lane - 16];
    }
}

extern "C" void kernel_launch(void* const* d_in, const int* in_sizes, int n_in,
                              void* d_out, int out_size, void* d_ws, size_t ws_size,
                              hipStream_t stream) {
    (void)in_sizes; (void)n_in; (void)d_ws; (void)ws_size; (void)out_size;
    const float* quality   = (const float*)d_in[0];
    const float* frequency = (const float*)d_in[1];
    const float* hrss      = (const float*)d_in[2];
    const float* phase     = (const float*)d_in[3];
    const float* ecc       = (const float*)d_in[4];
    const float* shifts    = (const float*)d_in[5];
    // d_in[6] = times, computed analytically in-kernel

    float* outCross = (float*)d_out;                       // [G, T]
    float* outPlus  = (float*)d_out + G_CNT * T_LEN;       // [G, T]

    dim3 grid(T_LEN / T_PER_BLOCK, G_CNT);   // (128, 64)
    dim3 block(256);
    msg_wmma_kernel<<<grid, block, 0, stream>>>(quality, frequency, hrss, phase,
                                                ecc, shifts, outCross, outPlus);
}